// BCMAttention_57269093924944
// MI455X (gfx1250) — compile-verified
//
#include <hip/hip_runtime.h>
#include <hip/hip_bf16.h>
#include <math.h>

typedef float v2f __attribute__((ext_vector_type(2)));
typedef float v8f __attribute__((ext_vector_type(8)));

#define NB 8
#define NC 128
#define NP 2000
#define NH 4
#define KNN 9
#define BCN (NB*NC*NP)          /* 2,048,000 */
#define H1COLS 6000             /* N*3 columns of the stride-3 conv1 output */

// ---- fp32 WMMA: D = A(16x4) * B(4x16) + C(16x16), exact fp32 ----
__device__ __forceinline__ v8f wmma4(v2f a, v2f b, v8f c) {
  return __builtin_amdgcn_wmma_f32_16x16x4_f32(false, a, false, b, (short)0, c,
                                               false, false);
}

// =====================================================================
// Kernel 1: q/k/v 1x1 filter convs.  out[f][b][co][n] = W[co][ci]*x[b][ci][n]+bias
// Grid (125, B, 3), block 256 (8 waves -> co tiles 0..7), 16 cols per block.
// =====================================================================
__global__ void __launch_bounds__(256) k_conv1x1(
    const float* __restrict__ x,
    const float* __restrict__ w0, const float* __restrict__ w1, const float* __restrict__ w2,
    const float* __restrict__ bb0, const float* __restrict__ bb1, const float* __restrict__ bb2,
    float* __restrict__ out)
{
  const int f = blockIdx.z, b = blockIdx.y, n0 = blockIdx.x * 16;
  const float* W    = (f == 0) ? w0  : (f == 1) ? w1  : w2;
  const float* bias = (f == 0) ? bb0 : (f == 1) ? bb1 : bb2;
  __shared__ float Bs[128][16];
  const int tid = threadIdx.x;
  for (int e = tid; e < 128 * 16; e += 256) {
    int k = e >> 4, c = e & 15;
    Bs[k][c] = x[((b * NC) + k) * NP + n0 + c];
  }
  __syncthreads();
  const int lane = tid & 31, wave = tid >> 5;
  const int co0 = wave * 16;
  const int m = lane & 15, ksel = (lane >> 4) << 1;
  v8f acc = {};
  for (int k0 = 0; k0 < 128; k0 += 4) {
    v2f a, bf;
    a.x  = W[(co0 + m) * 128 + k0 + ksel];
    a.y  = W[(co0 + m) * 128 + k0 + ksel + 1];
    bf.x = Bs[k0 + ksel][m];
    bf.y = Bs[k0 + ksel + 1][m];
    acc = wmma4(a, bf, acc);
  }
  float* outf = out + (size_t)f * BCN;
  for (int i = 0; i < 8; ++i) {
    int co = co0 + i + ((lane >> 4) << 3);
    outf[((b * NC) + co) * NP + n0 + m] = acc[i] + bias[co];
  }
}

// =====================================================================
// Kernel 2: instance-norm stats per (f,b,c) over N.
// =====================================================================
__global__ void __launch_bounds__(256) k_stats(
    const float* __restrict__ buf, float* __restrict__ meanW, float* __restrict__ rstdW)
{
  const int id = blockIdx.x;              // f*NB*NC + b*NC + c
  const float* p = buf + (size_t)id * NP;
  __shared__ float r1[256], r2[256];
  const int tid = threadIdx.x;
  float s1 = 0.f, s2 = 0.f;
  for (int n = tid; n < NP; n += 256) { float v = p[n]; s1 += v; s2 += v * v; }
  r1[tid] = s1; r2[tid] = s2;
  __syncthreads();
  for (int s = 128; s > 0; s >>= 1) {
    if (tid < s) { r1[tid] += r1[tid + s]; r2[tid] += r2[tid + s]; }
    __syncthreads();
  }
  if (tid == 0) {
    float mu = r1[0] * (1.f / NP);
    meanW[id] = mu;
    rstdW[id] = rsqrtf(r2[0] * (1.f / NP) - mu * mu + 1e-3f);
  }
}

// =====================================================================
// Kernel 3: apply IN + BN(affine) + relu in place, and write transposed
// copy tT[f][b][n][c] (tiled transpose via LDS for coalescing).
// Grid (63, NB*4, 3), block 256; tile = 32c x 32n.
// =====================================================================
__global__ void __launch_bounds__(256) k_norm(
    float* __restrict__ buf, float* __restrict__ tT,
    const float* __restrict__ g0, const float* __restrict__ g1, const float* __restrict__ g2,
    const float* __restrict__ e0, const float* __restrict__ e1, const float* __restrict__ e2,
    const float* __restrict__ meanW, const float* __restrict__ rstdW)
{
  const int f = blockIdx.z;
  const int b = blockIdx.y >> 2, ct = blockIdx.y & 3;
  const int nt = blockIdx.x * 32;
  const float* g  = (f == 0) ? g0 : (f == 1) ? g1 : g2;
  const float* be = (f == 0) ? e0 : (f == 1) ? e1 : e2;
  const float bnr = rsqrtf(1.0f + 1e-5f);
  __shared__ float tile[32][33];
  const int tid = threadIdx.x;
  const int cr = tid >> 3, ngb = (tid & 7) * 4;
  const int c = ct * 32 + cr;
  const int sid = (f * NB + b) * NC + c;
  const float mu = meanW[sid], rs = rstdW[sid];
  const float sc = g[c] * bnr, bo = be[c];
  float* bp = buf + (size_t)f * BCN + ((size_t)(b * NC) + c) * NP;
  for (int q = 0; q < 4; ++q) {
    int n = nt + ngb + q;
    float v = 0.f;
    if (n < NP) {
      v = (bp[n] - mu) * rs * sc + bo;
      v = fmaxf(v, 0.f);
      bp[n] = v;
    }
    tile[cr][ngb + q] = v;
  }
  __syncthreads();
  const int nr = tid >> 3, cgb = (tid & 7) * 4;
  const int n = nt + nr;
  if (n < NP) {
    float* tp = tT + (size_t)f * BCN + ((size_t)(b * NP) + n) * NC + ct * 32;
    for (int q = 0; q < 4; ++q) tp[cgb + q] = tile[cgb + q][nr];
  }
}

// =====================================================================
// Kernel 4: row squared norms xx[f][b][n] from transposed layout.
// =====================================================================
__global__ void __launch_bounds__(256) k_xx(
    const float* __restrict__ tT, float* __restrict__ xxW)
{
  const int t = blockIdx.x * 256 + threadIdx.x;
  if (t >= 3 * NB * NP) return;
  const float* p = tT + (size_t)t * NC;
  float s = 0.f;
  for (int c = 0; c < NC; ++c) { float v = p[c]; s += v * v; }
  xxW[t] = s;
}

// =====================================================================
// Kernel 5: fused gram + top-9 KNN.
// Block = 128 threads (4 waves), owns 32 rows; scans 63 column tiles of 32.
// pd = 2*<x_r,x_m> - xx_r - xx_m ; gram tiles via fp32 WMMA from LDS.
// =====================================================================
__global__ void __launch_bounds__(128) k_knn(
    const float* __restrict__ buf, const float* __restrict__ xxW, int* __restrict__ idxW)
{
  const int f = blockIdx.z, b = blockIdx.y;
  const int row0 = blockIdx.x * 32;
  const float* X  = buf + (size_t)f * BCN + (size_t)b * NC * NP;   // [c][n]
  const float* xx = xxW + (f * NB + b) * NP;
  __shared__ float As[128][32];
  __shared__ float Bs[128][32];
  __shared__ float P[32][33];
  __shared__ float xr[32], xc[32];
  const int tid = threadIdx.x;
  for (int e = tid; e < 128 * 32; e += 128) {
    int c = e >> 5, r = e & 31;
    int gr = row0 + r;
    As[c][r] = (gr < NP) ? X[c * NP + gr] : 0.f;
  }
  if (tid < 32) xr[tid] = (row0 + tid < NP) ? xx[row0 + tid] : 0.f;

  float tv[9]; int ti[9];
  for (int i = 0; i < 9; ++i) { tv[i] = -3.4e38f; ti[i] = 0; }

  const int lane = tid & 31, wave = tid >> 5;
  const int rh = (wave >> 1) * 16, ch = (wave & 1) * 16;
  const int m = lane & 15, ksel = (lane >> 4) << 1;

  for (int ctile = 0; ctile < (NP + 31) / 32; ++ctile) {
    const int col0 = ctile * 32;
    __syncthreads();                         // previous scan done; safe to refill Bs
    for (int e = tid; e < 128 * 32; e += 128) {
      int c = e >> 5, r = e & 31;
      int gc = col0 + r;
      Bs[c][r] = (gc < NP) ? X[c * NP + gc] : 0.f;
    }
    if (tid < 32) xc[tid] = (col0 + tid < NP) ? xx[col0 + tid] : 0.f;
    __syncthreads();                         // tiles ready
    v8f acc = {};
    for (int k0 = 0; k0 < 128; k0 += 4) {
      v2f a, bf;
      a.x  = As[k0 + ksel][rh + m];
      a.y  = As[k0 + ksel + 1][rh + m];
      bf.x = Bs[k0 + ksel][ch + m];
      bf.y = Bs[k0 + ksel + 1][ch + m];
      acc = wmma4(a, bf, acc);
    }
    for (int i = 0; i < 8; ++i) {
      int r = rh + i + ((lane >> 4) << 3);
      int c = ch + m;
      P[r][c] = 2.f * acc[i] - xr[r] - xc[c];
    }
    __syncthreads();                         // P ready
    if (tid < 32) {
      for (int j = 0; j < 32; ++j) {
        int mm = col0 + j;
        if (mm < NP) {
          float val = P[tid][j];
          if (val > tv[8]) {                 // strict > keeps lower-index ties
            int p = 8;
            while (p > 0 && val > tv[p - 1]) {
              tv[p] = tv[p - 1]; ti[p] = ti[p - 1]; --p;
            }
            tv[p] = val; ti[p] = mm;
          }
        }
      }
    }
  }
  if (tid < 32 && row0 + tid < NP) {
    int* op = idxW + ((size_t)(f * NB + b) * NP + row0 + tid) * KNN;
    for (int i = 0; i < 9; ++i) op[i] = ti[i];
  }
}

// =====================================================================
// Kernel 6: gather-fused DGCNN conv1 (kernel (1,3), stride 3) + BN + relu.
// Output columns col = n*3+j  (6000 per batch). K = 3*256 = 768 with
// edge features [ctr ; ctr - nbr] built directly into LDS.
// Grid (375, B, 3), block 256 (8 waves cover co 0..127).
// =====================================================================
__global__ void __launch_bounds__(256) k_gconv1(
    const float* __restrict__ tT, const int* __restrict__ idxW,
    const float* __restrict__ w1a, const float* __restrict__ w1b, const float* __restrict__ w1c,
    const float* __restrict__ b1a, const float* __restrict__ b1b, const float* __restrict__ b1c,
    const float* __restrict__ g1a, const float* __restrict__ g1b, const float* __restrict__ g1c,
    const float* __restrict__ e1a, const float* __restrict__ e1b, const float* __restrict__ e1c,
    float* __restrict__ h1)
{
  const int f = blockIdx.z, b = blockIdx.y, col0 = blockIdx.x * 16;
  const float* W  = (f == 0) ? w1a : (f == 1) ? w1b : w1c;
  const float* bs = (f == 0) ? b1a : (f == 1) ? b1b : b1c;
  const float* gs = (f == 0) ? g1a : (f == 1) ? g1b : g1c;
  const float* es = (f == 0) ? e1a : (f == 1) ? e1b : e1c;
  const float* T   = tT + (size_t)f * BCN + (size_t)b * NP * NC;     // [n][c]
  const int*   idx = idxW + (size_t)(f * NB + b) * NP * KNN;
  __shared__ float Bs[768][16];
  __shared__ int nOf[16], nbr[16][3];
  const int tid = threadIdx.x;
  if (tid < 16) {
    int col = col0 + tid;
    int n = col / 3, jj = col % 3;
    nOf[tid] = n;
    for (int kw = 0; kw < 3; ++kw) nbr[tid][kw] = idx[n * KNN + jj * 3 + kw];
  }
  __syncthreads();
  for (int e = tid; e < 768 * 16; e += 256) {
    int cc = e / 768, k = e - cc * 768;
    int kw = k >> 8, ci = k & 255;           // K index = kw*256 + ci
    int n = nOf[cc];
    float ctr = T[n * NC + (ci & 127)];
    float v = ctr;
    if (ci >= 128) v = ctr - T[nbr[cc][kw] * NC + (ci - 128)];
    Bs[k][cc] = v;
  }
  __syncthreads();
  const int lane = tid & 31, wave = tid >> 5;
  const int co0 = wave * 16;
  const int m = lane & 15, ksel = (lane >> 4) << 1;
  v8f acc = {};
  for (int k0 = 0; k0 < 768; k0 += 4) {
    int ka = k0 + ksel, kb = ka + 1;
    v2f a, bf;
    // w1 layout [co][ci][kw] (kw fastest); our k = kw*256 + ci
    a.x  = W[(co0 + m) * 768 + (ka & 255) * 3 + (ka >> 8)];
    a.y  = W[(co0 + m) * 768 + (kb & 255) * 3 + (kb >> 8)];
    bf.x = Bs[ka][m];
    bf.y = Bs[kb][m];
    acc = wmma4(a, bf, acc);
  }
  const float bnr = rsqrtf(1.f + 1e-5f);
  float* H = h1 + (size_t)(f * NB + b) * NC * H1COLS;
  for (int i = 0; i < 8; ++i) {
    int co = co0 + i + ((lane >> 4) << 3);
    float v = (acc[i] + bs[co]) * (gs[co] * bnr) + es[co];
    H[co * H1COLS + col0 + m] = fmaxf(v, 0.f);
  }
}

// =====================================================================
// Kernel 7: DGCNN conv2 (kernel (1,3)) + BN + relu -> h2[f][b][co][n].
// K = 128*3 = 384. Grid (125, B, 3), block 256.
// =====================================================================
__global__ void __launch_bounds__(256) k_gconv2(
    const float* __restrict__ h1,
    const float* __restrict__ w2a, const float* __restrict__ w2b, const float* __restrict__ w2c,
    const float* __restrict__ b2a, const float* __restrict__ b2b, const float* __restrict__ b2c,
    const float* __restrict__ g2a, const float* __restrict__ g2b, const float* __restrict__ g2c,
    const float* __restrict__ e2a, const float* __restrict__ e2b, const float* __restrict__ e2c,
    float* __restrict__ h2)
{
  const int f = blockIdx.z, b = blockIdx.y, n0 = blockIdx.x * 16;
  const float* W  = (f == 0) ? w2a : (f == 1) ? w2b : w2c;
  const float* bs = (f == 0) ? b2a : (f == 1) ? b2b : b2c;
  const float* gs = (f == 0) ? g2a : (f == 1) ? g2b : g2c;
  const float* es = (f == 0) ? e2a : (f == 1) ? e2b : e2c;
  const float* H = h1 + (size_t)(f * NB + b) * NC * H1COLS;
  __shared__ float Bs[384][16];
  const int tid = threadIdx.x;
  for (int e = tid; e < 384 * 16; e += 256) {
    int ci = e / 48, r = e - ci * 48;        // r = cc*3 + kw, contiguous in h1
    Bs[ci * 3 + (r % 3)][r / 3] = H[ci * H1COLS + n0 * 3 + r];
  }
  __syncthreads();
  const int lane = tid & 31, wave = tid >> 5;
  const int co0 = wave * 16;
  const int m = lane & 15, ksel = (lane >> 4) << 1;
  v8f acc = {};
  for (int k0 = 0; k0 < 384; k0 += 4) {
    v2f a, bf;
    a.x  = W[(co0 + m) * 384 + k0 + ksel];       // w2: [co][ci][kw] == k directly
    a.y  = W[(co0 + m) * 384 + k0 + ksel + 1];
    bf.x = Bs[k0 + ksel][m];
    bf.y = Bs[k0 + ksel + 1][m];
    acc = wmma4(a, bf, acc);
  }
  const float bnr = rsqrtf(1.f + 1e-5f);
  float* O = h2 + (size_t)f * BCN;
  for (int i = 0; i < 8; ++i) {
    int co = co0 + i + ((lane >> 4) << 3);
    float v = (acc[i] + bs[co]) * (gs[co] * bnr) + es[co];
    O[((b * NC) + co) * NP + n0 + m] = fmaxf(v, 0.f);
  }
}

// =====================================================================
// Kernel 8: attention per (b, head): L2-norms over N, 32x32 logits,
// softmax, apply to V.  Small FLOPs -> VALU.
// =====================================================================
__global__ void __launch_bounds__(256) k_attn(
    const float* __restrict__ h2, const float* __restrict__ temp, float* __restrict__ oAttn)
{
  const int b = blockIdx.x >> 2, h = blockIdx.x & 3;
  const float* Qb = h2 + (size_t)(b * NC + h * 32) * NP;
  const float* Kb = Qb + (size_t)BCN;
  const float* Vb = Qb + 2ull * (size_t)BCN;
  __shared__ float red[64][4];
  __shared__ float norms[64];
  __shared__ float attn[32][32];
  const int tid = threadIdx.x;
  { // row norms: 64 rows (32 q + 32 k), 4 partials each
    int r = tid >> 2, part = tid & 3;
    const float* base = (r < 32) ? (Qb + r * NP) : (Kb + (r - 32) * NP);
    float s = 0.f;
    for (int n = part; n < NP; n += 4) { float v = base[n]; s += v * v; }
    red[r][part] = s;
  }
  __syncthreads();
  if (tid < 64) norms[tid] = sqrtf(red[tid][0] + red[tid][1] + red[tid][2] + red[tid][3]);
  __syncthreads();
  const float tp = temp[h];
  for (int p = tid; p < 1024; p += 256) {
    int i = p >> 5, j = p & 31;
    const float* qi = Qb + i * NP;
    const float* kj = Kb + j * NP;
    float d = 0.f;
    for (int n = 0; n < NP; ++n) d += qi[n] * kj[n];
    attn[i][j] = d * tp / (fmaxf(norms[i], 1e-12f) * fmaxf(norms[32 + j], 1e-12f));
  }
  __syncthreads();
  if (tid < 32) {
    float mx = -3.4e38f;
    for (int j = 0; j < 32; ++j) mx = fmaxf(mx, attn[tid][j]);
    float ev[32], sm = 0.f;
    for (int j = 0; j < 32; ++j) { ev[j] = expf(attn[tid][j] - mx); sm += ev[j]; }
    float inv = 1.f / sm;
    for (int j = 0; j < 32; ++j) attn[tid][j] = ev[j] * inv;
  }
  __syncthreads();
  float* O = oAttn + (size_t)(b * NC + h * 32) * NP;
  for (int n = tid; n < NP; n += 256) {
    float o[32];
    for (int i = 0; i < 32; ++i) o[i] = 0.f;
    for (int j = 0; j < 32; ++j) {
      float vj = Vb[j * NP + n];
      for (int i = 0; i < 32; ++i) o[i] += attn[i][j] * vj;
    }
    for (int i = 0; i < 32; ++i) O[i * NP + n] = o[i];
  }
}

// =====================================================================
// Kernel 9: final 1x1 conv + bias + residual -> d_out.
// =====================================================================
__global__ void __launch_bounds__(256) k_final(
    const float* __restrict__ oAttn, const float* __restrict__ x,
    const float* __restrict__ W, const float* __restrict__ bias, float* __restrict__ out)
{
  const int b = blockIdx.y, n0 = blockIdx.x * 16;
  __shared__ float Bs[128][16];
  const int tid = threadIdx.x;
  for (int e = tid; e < 128 * 16; e += 256) {
    int k = e >> 4, c = e & 15;
    Bs[k][c] = oAttn[((b * NC) + k) * NP + n0 + c];
  }
  __syncthreads();
  const int lane = tid & 31, wave = tid >> 5;
  const int co0 = wave * 16;
  const int m = lane & 15, ksel = (lane >> 4) << 1;
  v8f acc = {};
  for (int k0 = 0; k0 < 128; k0 += 4) {
    v2f a, bf;
    a.x  = W[(co0 + m) * 128 + k0 + ksel];
    a.y  = W[(co0 + m) * 128 + k0 + ksel + 1];
    bf.x = Bs[k0 + ksel][m];
    bf.y = Bs[k0 + ksel + 1][m];
    acc = wmma4(a, bf, acc);
  }
  for (int i = 0; i < 8; ++i) {
    int co = co0 + i + ((lane >> 4) << 3);
    size_t o = (size_t)((b * NC) + co) * NP + n0 + m;
    out[o] = acc[i] + bias[co] + x[o];
  }
}

// =====================================================================
extern "C" void kernel_launch(void* const* d_in, const int* in_sizes, int n_in,
                              void* d_out, int out_size, void* d_ws, size_t ws_size,
                              hipStream_t stream)
{
  (void)in_sizes; (void)n_in; (void)out_size; (void)ws_size;
  // Input order follows setup_inputs() dict (recursive insertion order):
  // 0: x ; 1..12: qf/kf/vf {w,b,bn_g,bn_b} ; 13..36: gq/gk/gv {w1,b1,g1,be1,w2,b2,g2,be2}
  // 37: temp ; 38: po_w ; 39: po_b
  const float* x = (const float*)d_in[0];
  const float *fw[3], *fb[3], *fg[3], *fbe[3];
  for (int f = 0; f < 3; ++f) {
    fw[f]  = (const float*)d_in[1 + f * 4 + 0];
    fb[f]  = (const float*)d_in[1 + f * 4 + 1];
    fg[f]  = (const float*)d_in[1 + f * 4 + 2];
    fbe[f] = (const float*)d_in[1 + f * 4 + 3];
  }
  const float *gw1[3], *gb1[3], *gg1[3], *ge1[3], *gw2[3], *gb2[3], *gg2[3], *ge2[3];
  for (int g = 0; g < 3; ++g) {
    const int base = 13 + g * 8;
    gw1[g] = (const float*)d_in[base + 0]; gb1[g] = (const float*)d_in[base + 1];
    gg1[g] = (const float*)d_in[base + 2]; ge1[g] = (const float*)d_in[base + 3];
    gw2[g] = (const float*)d_in[base + 4]; gb2[g] = (const float*)d_in[base + 5];
    gg2[g] = (const float*)d_in[base + 6]; ge2[g] = (const float*)d_in[base + 7];
  }
  const float* temp = (const float*)d_in[37];
  const float* po_w = (const float*)d_in[38];
  const float* po_b = (const float*)d_in[39];
  float* out = (float*)d_out;

  // Workspace layout (floats), ~158 MB total.
  float* wsf   = (float*)d_ws;
  const size_t bcn = (size_t)BCN;
  float* buf   = wsf;                         // [3][B][C][N] conv -> normalized in place
  float* tT    = wsf + 3 * bcn;               // [3][B][N][C]
  float* meanW = wsf + 6 * bcn;               // 3*B*C
  float* rstdW = meanW + 3 * NB * NC;         // 3*B*C
  float* xxW   = rstdW + 3 * NB * NC;         // [3][B][N]
  int*   idxW  = (int*)(xxW + 3 * NB * NP);   // [3][B][N][9]
  float* h1W   = (float*)(idxW + (size_t)3 * NB * NP * KNN);   // [3][B][C][6000]
  float* h2W   = h1W + (size_t)3 * NB * NC * H1COLS;           // [3][B][C][N]
  float* oAtt  = h2W + 3 * bcn;                                // [B][C][N]

  dim3 blk(256);
  k_conv1x1<<<dim3(NP / 16, NB, 3), blk, 0, stream>>>(
      x, fw[0], fw[1], fw[2], fb[0], fb[1], fb[2], buf);
  k_stats<<<dim3(3 * NB * NC), blk, 0, stream>>>(buf, meanW, rstdW);
  k_norm<<<dim3((NP + 31) / 32, NB * 4, 3), blk, 0, stream>>>(
      buf, tT, fg[0], fg[1], fg[2], fbe[0], fbe[1], fbe[2], meanW, rstdW);
  k_xx<<<dim3((3 * NB * NP + 255) / 256), blk, 0, stream>>>(tT, xxW);
  k_knn<<<dim3((NP + 31) / 32, NB, 3), dim3(128), 0, stream>>>(buf, xxW, idxW);
  k_gconv1<<<dim3(H1COLS / 16, NB, 3), blk, 0, stream>>>(
      tT, idxW,
      gw1[0], gw1[1], gw1[2], gb1[0], gb1[1], gb1[2],
      gg1[0], gg1[1], gg1[2], ge1[0], ge1[1], ge1[2], h1W);
  k_gconv2<<<dim3(NP / 16, NB, 3), blk, 0, stream>>>(
      h1W,
      gw2[0], gw2[1], gw2[2], gb2[0], gb2[1], gb2[2],
      gg2[0], gg2[1], gg2[2], ge2[0], ge2[1], ge2[2], h2W);
  k_attn<<<dim3(NB * NH), blk, 0, stream>>>(h2W, temp, oAtt);
  k_final<<<dim3(NP / 16, NB), blk, 0, stream>>>(oAtt, x, po_w, po_b, out);
}